// GATlayerV2_45105746542632
// MI455X (gfx1250) — compile-verified
//
#include <hip/hip_runtime.h>
#include <math.h>

#define NN 100000
#define EE 1600000
#define DD 128
#define LEAKY 0.2f

typedef float v2f __attribute__((ext_vector_type(2)));
typedef float v8f __attribute__((ext_vector_type(8)));

__device__ __forceinline__ v8f wmma4(v2f a, v2f b, v8f c) {
    // V_WMMA_F32_16X16X4_F32 : full-precision f32 matrix op (CDNA5)
    return __builtin_amdgcn_wmma_f32_16x16x4_f32(
        /*neg_a=*/false, a, /*neg_b=*/false, b,
        /*c_mod=*/(short)0, c, /*reuse_a=*/false, /*reuse_b=*/false);
}

// monotonic float<->uint mapping for atomic segment-max
__device__ __forceinline__ unsigned enc_f(float f) {
    unsigned u = __float_as_uint(f);
    return (u & 0x80000000u) ? ~u : (u | 0x80000000u);
}
__device__ __forceinline__ float dec_f(unsigned u) {
    unsigned v = (u & 0x80000000u) ? (u & 0x7FFFFFFFu) : ~u;
    return __uint_as_float(v);
}

__device__ __forceinline__ float sigmoidf_(float x) { return 1.0f / (1.0f + expf(-x)); }

// ---------------------------------------------------------------------------
// 1) init: zero transform/denom/deg, seed m = enc(-inf)
// ---------------------------------------------------------------------------
__global__ __launch_bounds__(256) void init_kernel(float* __restrict__ transform,
                                                   unsigned* __restrict__ mmax,
                                                   float* __restrict__ denom,
                                                   float* __restrict__ deg) {
    int i = blockIdx.x * 256 + threadIdx.x;
    if (i < NN * DD) transform[i] = 0.0f;
    if (i < NN) {
        mmax[i]  = 0x007FFFFFu;   // enc(-inf)
        denom[i] = 0.0f;
        deg[i]   = 0.0f;
    }
}

// ---------------------------------------------------------------------------
// 2) nft = A @ W2^T + b2   (WMMA f32 16x16x4, A tile staged in LDS)
//    grid.x = NN/16, block = 256 (8 waves -> 8 column tiles covering 128 cols)
// ---------------------------------------------------------------------------
__global__ __launch_bounds__(256) void nft_gemm_kernel(const float* __restrict__ A,
                                                       const float* __restrict__ W2,
                                                       const float* __restrict__ b2,
                                                       float* __restrict__ nft) {
    __shared__ float a_s[16 * DD];
    const int row0 = blockIdx.x * 16;
    const int tid  = threadIdx.x;

    // stage 16x128 A tile (512 float4, 2 per thread)
    const float4* Ab  = (const float4*)(A + (size_t)row0 * DD);
    float4*       as4 = (float4*)a_s;
    as4[tid]       = Ab[tid];
    as4[tid + 256] = Ab[tid + 256];
    __syncthreads();

    const int wave = tid >> 5;
    const int lane = tid & 31;
    const int n0   = wave * 16;
    const int lmod = lane & 15;
    const int koff = (lane >> 4) << 1;  // 0 or 2

    const float* Ar = a_s + lmod * DD;                 // A row fragment source
    const float* Wr = W2 + (size_t)(n0 + lmod) * DD;   // B column n = W2 row n

    v8f acc = {0.f, 0.f, 0.f, 0.f, 0.f, 0.f, 0.f, 0.f};
#pragma unroll
    for (int kb = 0; kb < DD; kb += 4) {
        v2f a = *(const v2f*)(Ar + kb + koff);
        v2f b = *(const v2f*)(Wr + kb + koff);
        acc = wmma4(a, b, acc);
    }

    const int col  = n0 + lmod;
    const float bb = b2[col];
    const int rb   = row0 + ((lane >> 4) << 3);
#pragma unroll
    for (int v = 0; v < 8; ++v)
        nft[(size_t)(rb + v) * DD + col] = acc[v] + bb;
}

// ---------------------------------------------------------------------------
// 3) logit / logit_src : wave-per-node dual dot product (wave32 shuffle reduce)
// ---------------------------------------------------------------------------
__global__ __launch_bounds__(256) void logits_kernel(const float* __restrict__ A,
                                                     const float* __restrict__ W1,
                                                     float* __restrict__ logit,
                                                     float* __restrict__ logit_src) {
    const int node = blockIdx.x * 8 + (threadIdx.x >> 5);
    const int lane = threadIdx.x & 31;
    if (node >= NN) return;
    float4 a  = ((const float4*)(A + (size_t)node * DD))[lane];
    float4 wd = ((const float4*)W1)[lane];          // W1[0][0:128]
    float4 ws = ((const float4*)(W1 + DD))[lane];   // W1[0][128:256]
    float pd = a.x * wd.x + a.y * wd.y + a.z * wd.z + a.w * wd.w;
    float ps = a.x * ws.x + a.y * ws.y + a.z * ws.z + a.w * ws.w;
#pragma unroll
    for (int off = 16; off > 0; off >>= 1) {
        pd += __shfl_xor(pd, off, 32);
        ps += __shfl_xor(ps, off, 32);
    }
    if (lane == 0) {
        logit[node]     = pd;
        logit_src[node] = ps;
    }
}

// ---------------------------------------------------------------------------
// 4) per-edge score + leaky ReLU + segment-max (atomic on ordered uint) + deg
// ---------------------------------------------------------------------------
__global__ __launch_bounds__(256) void edge_score_kernel(const int* __restrict__ ei,
                                                         const float* __restrict__ logit,
                                                         const float* __restrict__ logit_src,
                                                         const float* __restrict__ b1,
                                                         float* __restrict__ score,
                                                         unsigned* __restrict__ mmax,
                                                         float* __restrict__ deg) {
    const int e = blockIdx.x * 256 + threadIdx.x;
    if (e >= EE) return;
    const int s = ei[e];
    const int d = ei[EE + e];
    float sc = logit[d] + logit_src[s] + b1[0];
    sc = (sc > 0.0f) ? sc : LEAKY * sc;
    score[e] = sc;
    atomicMax(&mmax[d], enc_f(sc));
    atomicAdd(&deg[d], 1.0f);
}

// ---------------------------------------------------------------------------
// 5) e = exp(score - m[dst]); denom[dst] += e   (e overwrites score buffer)
// ---------------------------------------------------------------------------
__global__ __launch_bounds__(256) void edge_exp_kernel(const int* __restrict__ ei,
                                                       const unsigned* __restrict__ mmax,
                                                       float* __restrict__ score,
                                                       float* __restrict__ denom) {
    const int e = blockIdx.x * 256 + threadIdx.x;
    if (e >= EE) return;
    const int d  = ei[EE + e];
    const float m  = dec_f(mmax[d]);     // finite: d has >=1 incoming edge
    const float ev = expf(score[e] - m);
    score[e] = ev;
    atomicAdd(&denom[d], ev);
}

// ---------------------------------------------------------------------------
// 6) aggregation: one wave per edge, float4 gather of nft[src], f32 atomics
// ---------------------------------------------------------------------------
__global__ __launch_bounds__(256) void edge_agg_kernel(const int* __restrict__ ei,
                                                       const float* __restrict__ evals,
                                                       const float* __restrict__ denom,
                                                       const float* __restrict__ nft,
                                                       float* __restrict__ transform) {
    const int e = blockIdx.x * 8 + (threadIdx.x >> 5);
    if (e >= EE) return;
    const int lane = threadIdx.x & 31;
    const int s = ei[e];
    const int d = ei[EE + e];
    const float attn = evals[e] / denom[d];
    float4 v = ((const float4*)(nft + (size_t)s * DD))[lane];
    float* t = transform + (size_t)d * DD + lane * 4;
    atomicAdd(t + 0, attn * v.x);
    atomicAdd(t + 1, attn * v.y);
    atomicAdd(t + 2, attn * v.z);
    atomicAdd(t + 3, attn * v.w);
}

// ---------------------------------------------------------------------------
// 7) out_emb = deg>0 ? ELU(transform) : A      (in-place in transform)
// ---------------------------------------------------------------------------
__global__ __launch_bounds__(256) void finalize_kernel(const float* __restrict__ A,
                                                       const float* __restrict__ deg,
                                                       float* __restrict__ transform) {
    const int i = blockIdx.x * 256 + threadIdx.x;
    if (i >= NN * DD) return;
    const int node = i >> 7;
    const float t   = transform[i];
    const float ctx = (t > 0.0f) ? t : (expf(fminf(t, 0.0f)) - 1.0f);  // ELU alpha=1
    transform[i] = (deg[node] > 0.0f) ? ctx : A[i];
}

// ---------------------------------------------------------------------------
// 8) fused GRU: gi = out_emb@w_ih^T, gh = A@w_hh^T via WMMA into LDS, gates.
//    grid.x = NN/16, block = 256; 48 output tiles (24 gi + 24 gh), 6 per wave.
// ---------------------------------------------------------------------------
__global__ __launch_bounds__(256) void gru_kernel(const float* __restrict__ out_emb,
                                                  const float* __restrict__ A,
                                                  const float* __restrict__ w_ih,
                                                  const float* __restrict__ w_hh,
                                                  const float* __restrict__ b_ih,
                                                  const float* __restrict__ b_hh,
                                                  float* __restrict__ h_new) {
    __shared__ float gi_s[16 * 384];
    __shared__ float gh_s[16 * 384];

    const int row0 = blockIdx.x * 16;
    const int tid  = threadIdx.x;
    const int wave = tid >> 5;
    const int lane = tid & 31;
    const int lmod = lane & 15;
    const int koff = (lane >> 4) << 1;

    for (int t = 0; t < 6; ++t) {
        const int tile = wave * 6 + t;          // 0..47, wave-uniform
        const bool isGI = (tile < 24);
        const int n0 = (isGI ? tile : (tile - 24)) * 16;
        const float* src = isGI ? out_emb : A;
        const float* Wt  = isGI ? w_ih : w_hh;
        float* sdst      = isGI ? gi_s : gh_s;

        const float* Ar = src + (size_t)(row0 + lmod) * DD;
        const float* Wr = Wt + (size_t)(n0 + lmod) * DD;

        v8f acc = {0.f, 0.f, 0.f, 0.f, 0.f, 0.f, 0.f, 0.f};
#pragma unroll
        for (int kb = 0; kb < DD; kb += 4) {
            v2f a = *(const v2f*)(Ar + kb + koff);
            v2f b = *(const v2f*)(Wr + kb + koff);
            acc = wmma4(a, b, acc);
        }
        const int col = n0 + lmod;
        const int rb  = (lane >> 4) << 3;
#pragma unroll
        for (int v = 0; v < 8; ++v)
            sdst[(rb + v) * 384 + col] = acc[v];
    }
    __syncthreads();

    // gate phase: 16 rows x 128 cols, 8 elements per thread
    const int row = tid >> 4;
    const int c0  = (tid & 15) * 8;
#pragma unroll
    for (int j = 0; j < 8; ++j) {
        const int c = c0 + j;
        const float gir = gi_s[row * 384 + c]       + b_ih[c];
        const float giz = gi_s[row * 384 + 128 + c] + b_ih[128 + c];
        const float gin = gi_s[row * 384 + 256 + c] + b_ih[256 + c];
        const float ghr = gh_s[row * 384 + c]       + b_hh[c];
        const float ghz = gh_s[row * 384 + 128 + c] + b_hh[128 + c];
        const float ghn = gh_s[row * 384 + 256 + c] + b_hh[256 + c];
        const float r = sigmoidf_(gir + ghr);
        const float z = sigmoidf_(giz + ghz);
        const float n = tanhf(gin + r * ghn);
        const float h = A[(size_t)(row0 + row) * DD + c];
        h_new[(size_t)(row0 + row) * DD + c] = (1.0f - z) * n + z * h;
    }
}

// ---------------------------------------------------------------------------
extern "C" void kernel_launch(void* const* d_in, const int* in_sizes, int n_in,
                              void* d_out, int out_size, void* d_ws, size_t ws_size,
                              hipStream_t stream) {
    const float* A    = (const float*)d_in[0];  // atom_features [N,128]
    const float* W1   = (const float*)d_in[1];  // [1,256]
    const float* b1   = (const float*)d_in[2];  // [1]
    const float* W2   = (const float*)d_in[3];  // [128,128]
    const float* b2   = (const float*)d_in[4];  // [128]
    const float* w_ih = (const float*)d_in[5];  // [384,128]
    const float* w_hh = (const float*)d_in[6];  // [384,128]
    const float* b_ih = (const float*)d_in[7];  // [384]
    const float* b_hh = (const float*)d_in[8];  // [384]
    const int*   ei   = (const int*)d_in[9];    // [2,E] flat: [0:E]=src, [E:2E]=dst

    float* out = (float*)d_out;                 // h_new [N,128]

    // workspace layout (floats), ~111 MB total
    float*    nft       = (float*)d_ws;
    float*    transform = nft + (size_t)NN * DD;
    float*    escore    = transform + (size_t)NN * DD;   // score, then e
    float*    logit     = escore + EE;
    float*    logit_src = logit + NN;
    float*    denom     = logit_src + NN;
    float*    deg       = denom + NN;
    unsigned* mmax      = (unsigned*)(deg + NN);

    const int elems = NN * DD;                  // 12,800,000 (divisible by 256)

    init_kernel<<<elems / 256, 256, 0, stream>>>(transform, mmax, denom, deg);
    nft_gemm_kernel<<<NN / 16, 256, 0, stream>>>(A, W2, b2, nft);
    logits_kernel<<<NN / 8, 256, 0, stream>>>(A, W1, logit, logit_src);
    edge_score_kernel<<<EE / 256, 256, 0, stream>>>(ei, logit, logit_src, b1,
                                                    escore, mmax, deg);
    edge_exp_kernel<<<EE / 256, 256, 0, stream>>>(ei, mmax, escore, denom);
    edge_agg_kernel<<<EE / 8, 256, 0, stream>>>(ei, escore, denom, nft, transform);
    finalize_kernel<<<elems / 256, 256, 0, stream>>>(A, deg, transform);
    gru_kernel<<<NN / 16, 256, 0, stream>>>(transform, A, w_ih, w_hh, b_ih, b_hh, out);
}